// RobustMambaDiagSSM_15934328668592
// MI455X (gfx1250) — compile-verified
//
#include <hip/hip_runtime.h>
#include <hip/hip_bf16.h>

// ---------------------------------------------------------------------------
// RobustMambaDiagSSM for MI455X (gfx1250), fp32 end-to-end.
//   [0] copy W_in,W_out -> Wcat (batched)
//   [1] G_z = W_z @ W_z^T                       (WMMA f32 GEMM, z=2)
//   [2] 10x: t_z = trace(P_z); P_z <- (P_z@P_z)/t_z^2   (matrix squaring)
//   [3] finalize: v = top eigvec from P=G^1024, sigma^2 = v^T G v
//   [4] u_scaled = (u @ W_in^T) * scal[0]       (WMMA GEMM)
//   [5] raw      =  u @ Wp^T + bp               (WMMA GEMM)
//   [6] elementwise a,b,c + 2x2 spectral clamp, in-place over raw
//   [7] chunked 3-pass scan over T (64 chunks x 64), yhat = c * z
//   [8] out = (yhat @ W_out^T) * scal[1]        (WMMA GEMM)
// GEMM uses async Global->LDS DMA (ASYNCcnt) + double-buffered LDS tiles.
// Workspace: ~306 MB.
// ---------------------------------------------------------------------------

typedef __attribute__((ext_vector_type(2))) float v2f;
typedef __attribute__((ext_vector_type(8))) float v8f;
typedef __attribute__((ext_vector_type(4))) int   v4i;

#define TILE_M 128
#define TILE_N 64
#define TILE_K 32
#define LDSTR 36   // 144B rows: 16B-aligned for b128 LDS writes, 8B for b64 reads

#if __has_builtin(__builtin_amdgcn_global_load_async_to_lds_b128) && \
    __has_builtin(__builtin_amdgcn_s_wait_asynccnt)
#define HAVE_ASYNC_LDS 1
typedef __attribute__((address_space(1))) v4i* g_v4i;   // global source (AS1)
typedef __attribute__((address_space(3))) v4i* l_v4i;   // LDS destination (AS3)
// builtin signature (from clang diagnostics): (int4 AS1*, int4 AS3*, imm, imm)
#define ASYNC_LD128(gsrc, ldst) \
    __builtin_amdgcn_global_load_async_to_lds_b128( \
        (g_v4i)(void*)(gsrc), (l_v4i)(void*)(ldst), 0, 0)
#endif

// C[M,N] = (A[M,K] @ B[N,K]^T) * s + bias[n]
// scale_mode: 0 -> s=1 ; 1 -> s=scale_ptr[0] ; 2 -> s=1/(scale_ptr[z]^2)
__global__ __launch_bounds__(256) void wmma_gemm_f32(
    const float* __restrict__ A, const float* __restrict__ B,
    float* __restrict__ C, int M, int N, int K,
    long strideAz, long strideBz, long strideCz,
    const float* __restrict__ scale_ptr, int scale_mode,
    const float* __restrict__ bias)
{
    __shared__ float As[2][TILE_M][LDSTR];
    __shared__ float Bs[2][TILE_N][LDSTR];

    const int z = blockIdx.z;
    A += (long)z * strideAz;
    B += (long)z * strideBz;
    C += (long)z * strideCz;

    const int bm  = blockIdx.y * TILE_M;
    const int bn  = blockIdx.x * TILE_N;
    const int tid = threadIdx.x;
    const int lane = tid & 31;
    const int wid  = tid >> 5;          // 8 waves: 4 (M) x 2 (N)
    const int wm   = (wid >> 1) * 32;
    const int wn   = (wid & 1) * 32;
    const int lo   = lane & 15;
    const int hi   = lane >> 4;         // K-half select per ISA A/B layout

    v8f acc[2][2];
    {
        v8f zr = {};
        acc[0][0] = zr; acc[0][1] = zr; acc[1][0] = zr; acc[1][1] = zr;
    }

    const int c4 = (tid & 7) * 4;       // 8 float4 columns per 32-float row
    const int rb = tid >> 3;            // 0..31

#if defined(HAVE_ASYNC_LDS)
    // 6 async DMA instructions per wave per tile (4 A-rows + 2 B-rows)
    auto issue_tile = [&](int buf, int k0) {
        #pragma unroll
        for (int i = 0; i < 4; ++i) {
            const int r = rb + 32 * i;
            ASYNC_LD128(A + (long)(bm + r) * K + k0 + c4, &As[buf][r][c4]);
        }
        #pragma unroll
        for (int i = 0; i < 2; ++i) {
            const int r = rb + 32 * i;
            ASYNC_LD128(B + (long)(bn + r) * K + k0 + c4, &Bs[buf][r][c4]);
        }
    };
    issue_tile(0, 0);
#endif

    for (int k0 = 0; k0 < K; k0 += TILE_K) {
        const int cb = (k0 / TILE_K) & 1;
#if defined(HAVE_ASYNC_LDS)
        if (k0 + TILE_K < K) {
            issue_tile(cb ^ 1, k0 + TILE_K);           // prefetch next tile
            __builtin_amdgcn_s_wait_asynccnt(6);       // current tile's 6 done
        } else {
            __builtin_amdgcn_s_wait_asynccnt(0);
        }
#else
        // fallback: batch all global loads before any LDS store
        float4 ra[4], rbv[2];
        #pragma unroll
        for (int i = 0; i < 4; ++i)
            ra[i] = *(const float4*)(A + (long)(bm + rb + 32 * i) * K + k0 + c4);
        #pragma unroll
        for (int i = 0; i < 2; ++i)
            rbv[i] = *(const float4*)(B + (long)(bn + rb + 32 * i) * K + k0 + c4);
        #pragma unroll
        for (int i = 0; i < 4; ++i) {
            const int r = rb + 32 * i;
            As[cb][r][c4+0]=ra[i].x; As[cb][r][c4+1]=ra[i].y;
            As[cb][r][c4+2]=ra[i].z; As[cb][r][c4+3]=ra[i].w;
        }
        #pragma unroll
        for (int i = 0; i < 2; ++i) {
            const int r = rb + 32 * i;
            Bs[cb][r][c4+0]=rbv[i].x; Bs[cb][r][c4+1]=rbv[i].y;
            Bs[cb][r][c4+2]=rbv[i].z; Bs[cb][r][c4+3]=rbv[i].w;
        }
#endif
        __syncthreads();

        #pragma unroll
        for (int kk = 0; kk < TILE_K; kk += 4) {
            const int ks = kk + 2 * hi;
            v2f aF[2], bF[2];
            #pragma unroll
            for (int mt = 0; mt < 2; ++mt) {
                aF[mt][0] = As[cb][wm + mt * 16 + lo][ks];
                aF[mt][1] = As[cb][wm + mt * 16 + lo][ks + 1];
            }
            #pragma unroll
            for (int nt = 0; nt < 2; ++nt) {
                bF[nt][0] = Bs[cb][wn + nt * 16 + lo][ks];
                bF[nt][1] = Bs[cb][wn + nt * 16 + lo][ks + 1];
            }
            #pragma unroll
            for (int mt = 0; mt < 2; ++mt)
                #pragma unroll
                for (int nt = 0; nt < 2; ++nt)
                    acc[mt][nt] = __builtin_amdgcn_wmma_f32_16x16x4_f32(
                        false, aF[mt], false, bF[nt],
                        (short)0, acc[mt][nt], false, false);
        }
        __syncthreads();   // all waves done with buf cb before it is re-filled
    }

    float s = 1.0f;
    if (scale_mode == 1)      s = scale_ptr[0];
    else if (scale_mode == 2) { float t = scale_ptr[z]; s = 1.0f / (t * t); }

    #pragma unroll
    for (int mt = 0; mt < 2; ++mt)
        #pragma unroll
        for (int nt = 0; nt < 2; ++nt) {
            const int col = bn + wn + nt * 16 + lo;
            const float bv = bias ? bias[col] : 0.0f;
            #pragma unroll
            for (int r = 0; r < 8; ++r) {
                const int row = bm + wm + mt * 16 + r + 8 * hi;
                C[(long)row * N + col] = acc[mt][nt][r] * s + bv;
            }
        }
}

// trace of a 1024x1024 matrix per batch z
__global__ __launch_bounds__(256) void trace_k(const float* __restrict__ P,
                                               float* __restrict__ t)
{
    const int z = blockIdx.x;
    const float* Pz = P + (long)z * 1024 * 1024;
    __shared__ float red[256];
    float s = 0.0f;
    for (int i = threadIdx.x; i < 1024; i += 256)
        s += Pz[(long)i * 1024 + i];
    red[threadIdx.x] = s;
    __syncthreads();
    for (int k = 128; k > 0; k >>= 1) {
        if (threadIdx.x < k) red[threadIdx.x] += red[threadIdx.x + k];
        __syncthreads();
    }
    if (threadIdx.x == 0) t[z] = red[0];
}

// v = normalize(P r); sigma^2 = v^T G v; scal[z] = post-scale factors.
__global__ __launch_bounds__(1024) void finalize_sigma(
    const float* __restrict__ P, const float* __restrict__ G,
    const float* __restrict__ log_gamma, float* __restrict__ scal)
{
    const int z = blockIdx.x;
    const float* Pz = P + (long)z * 1024 * 1024;
    const float* Gz = G + (long)z * 1024 * 1024;
    __shared__ float vsh[1024];
    __shared__ float red[1024];
    const int i = threadIdx.x;

    float w = 0.0f;
    for (int j = 0; j < 1024; ++j) {
        const float rj = 1.0f + 0.001f * (float)(j & 31);  // fixed probe vector
        w += Pz[(long)i * 1024 + j] * rj;
    }
    red[i] = w * w;
    __syncthreads();
    for (int s = 512; s > 0; s >>= 1) {
        if (i < s) red[i] += red[i + s];
        __syncthreads();
    }
    const float nrm = sqrtf(red[0] + 1e-30f);
    vsh[i] = w / nrm;
    __syncthreads();

    float y = 0.0f;
    for (int j = 0; j < 1024; ++j)
        y += Gz[(long)i * 1024 + j] * vsh[j];
    red[i] = y * vsh[i];
    __syncthreads();
    for (int s = 512; s > 0; s >>= 1) {
        if (i < s) red[i] += red[i + s];
        __syncthreads();
    }
    if (i == 0) {
        const float lam   = fmaxf(red[0], 0.0f);
        const float sigma = sqrtf(lam);
        float inv = 1.0f / fmaxf(sigma, 1.0f);
        if (z == 0) inv *= expf(log_gamma[0]);   // fold gamma into u_scaled
        scal[z] = inv;
    }
}

__device__ __forceinline__ float softplus_f(float x) {
    return x > 20.0f ? x : log1pf(expf(x));
}

// in-place over raw: [bt][0..N)=a, [N..2N)=b*u_scaled, [2N..3N)=c
__global__ __launch_bounds__(256) void elemwise_abc(
    float* __restrict__ raw, const float* __restrict__ us,
    const float* __restrict__ alpha_log, const float* __restrict__ delta_bias)
{
    const int N = 1024;
    const long idx = (long)blockIdx.x * 256 + threadIdx.x;
    const long bt = idx >> 10;
    const int  n  = (int)(idx & 1023);
    const long rb = bt * 3 * N;

    const float alpha = softplus_f(alpha_log[n]);
    const float delta = softplus_f(raw[rb + n] + delta_bias[n]);
    float a = fminf(expf(-delta * alpha), 1.0f - 1e-4f);
    float b = tanhf(raw[rb + N + n]);
    float c = tanhf(raw[rb + 2 * N + n]);

    const float p  = a * a + c * c;
    const float r  = b * b;
    const float q  = a * b;
    const float d0 = p - r;
    const float disc = d0 * d0 + 4.0f * q * q;
    const float lam  = 0.5f * (p + r + sqrtf(disc + 1e-12f));
    const float sg   = sqrtf(lam + 1e-12f);
    const float inv  = 1.0f / fmaxf(sg, 1.0f);
    a *= inv; b *= inv; c *= inv;

    raw[rb + n]         = a;
    raw[rb + N + n]     = b * us[bt * N + n];
    raw[rb + 2 * N + n] = c;
}

#define SCAN_T 4096
#define SCAN_CLEN 64
#define SCAN_NCHUNK 64

// per-chunk affine composite (A, B): z_out = A*z_in + B
__global__ __launch_bounds__(256) void scan_pass1(
    const float* __restrict__ abc, float* __restrict__ Ac, float* __restrict__ Bc)
{
    const int N = 1024;
    const int n = blockIdx.x * 256 + threadIdx.x;
    const int chunk = blockIdx.y;
    const int b = blockIdx.z;
    const float* base = abc + (long)(b * SCAN_T + chunk * SCAN_CLEN) * (3 * N);
    float A = 1.0f, Bv = 0.0f;
    for (int t = 0; t < SCAN_CLEN; ++t) {
        const long o = (long)t * 3 * N;
        const float a  = base[o + n];
        const float bu = base[o + N + n];
        Bv = a * Bv + bu;
        A *= a;
    }
    const long ci = (long)(b * SCAN_NCHUNK + chunk) * N + n;
    Ac[ci] = A; Bc[ci] = Bv;
}

// exclusive scan of chunk composites -> chunk-start states S
__global__ __launch_bounds__(256) void scan_pass2(
    const float* __restrict__ Ac, const float* __restrict__ Bc,
    float* __restrict__ S)
{
    const int N = 1024;
    const int idx = blockIdx.x * 256 + threadIdx.x;  // B*N = 4096
    const int b = idx >> 10;
    const int n = idx & 1023;
    float carry = 0.0f;
    for (int c = 0; c < SCAN_NCHUNK; ++c) {
        const long ci = (long)(b * SCAN_NCHUNK + c) * N + n;
        S[ci] = carry;
        carry = Ac[ci] * carry + Bc[ci];
    }
}

// yhat[t] = c[t] * z_before_t ; z = a*z + bu
__global__ __launch_bounds__(256) void scan_pass3(
    const float* __restrict__ abc, const float* __restrict__ S,
    float* __restrict__ yhat)
{
    const int N = 1024;
    const int n = blockIdx.x * 256 + threadIdx.x;
    const int chunk = blockIdx.y;
    const int b = blockIdx.z;
    const float* base = abc + (long)(b * SCAN_T + chunk * SCAN_CLEN) * (3 * N);
    float* ybase = yhat + (long)(b * SCAN_T + chunk * SCAN_CLEN) * N;
    float zv = S[(long)(b * SCAN_NCHUNK + chunk) * N + n];
    for (int t = 0; t < SCAN_CLEN; ++t) {
        const long o = (long)t * 3 * N;
        const float a  = base[o + n];
        const float bu = base[o + N + n];
        const float cc = base[o + 2 * N + n];
        ybase[(long)t * N + n] = cc * zv;
        zv = a * zv + bu;
    }
}

extern "C" void kernel_launch(void* const* d_in, const int* in_sizes, int n_in,
                              void* d_out, int out_size, void* d_ws, size_t ws_size,
                              hipStream_t stream)
{
    (void)in_sizes; (void)n_in; (void)out_size; (void)ws_size;

    const float* u          = (const float*)d_in[0];
    const float* W_in       = (const float*)d_in[1];
    const float* W_out      = (const float*)d_in[2];
    const float* Wp         = (const float*)d_in[3];
    const float* bp         = (const float*)d_in[4];
    const float* alpha_log  = (const float*)d_in[5];
    const float* delta_bias = (const float*)d_in[6];
    const float* log_gamma  = (const float*)d_in[7];
    float* out = (float*)d_out;

    const int Bb = 4, T = 4096, D = 1024, Nn = 1024;
    const int BT = Bb * T;
    const long MAT = (long)1024 * 1024;        // elements per 1024^2 matrix

    // ---- workspace layout (~306 MB) ----
    char* ws = (char*)d_ws;
    const long RAW_B  = (long)BT * 3 * Nn * 4; // 192 MB
    const long US_B   = (long)BT * Nn * 4;     //  64 MB (u_scaled, reused as yhat)
    const long MAT_B  = MAT * 4;               //   4 MB
    const long CH_B   = (long)Bb * SCAN_NCHUNK * Nn * 4; // 1 MB
    long off = 0;
    float* raw  = (float*)(ws + off); off += RAW_B;
    float* us   = (float*)(ws + off); off += US_B;   // also yhat
    float* Wcat = (float*)(ws + off); off += 2 * MAT_B;
    float* Gcat = (float*)(ws + off); off += 2 * MAT_B;
    float* P0   = (float*)(ws + off); off += 2 * MAT_B;
    float* P1   = (float*)(ws + off); off += 2 * MAT_B;
    float* Ac   = (float*)(ws + off); off += CH_B;
    float* Bc   = (float*)(ws + off); off += CH_B;
    float* Sst  = (float*)(ws + off); off += CH_B;
    float* scal = (float*)(ws + off); off += 256;   // scal[0..1]
    float* tr   = scal + 8;                          // tr[0..1]

    // ---- [0] pack W_in / W_out contiguously for batched spectral pipeline ----
    (void)hipMemcpyAsync(Wcat,       W_in,  MAT_B, hipMemcpyDeviceToDevice, stream);
    (void)hipMemcpyAsync(Wcat + MAT, W_out, MAT_B, hipMemcpyDeviceToDevice, stream);

    // ---- [1] G_z = W_z @ W_z^T ----
    wmma_gemm_f32<<<dim3(1024 / TILE_N, 1024 / TILE_M, 2), 256, 0, stream>>>(
        Wcat, Wcat, Gcat, 1024, 1024, 1024, MAT, MAT, MAT, nullptr, 0, nullptr);

    // ---- [2] 10 trace-normalized squarings -> P = G^1024 (up to scale) ----
    const float* cur = Gcat;
    float* bufs[2] = {P0, P1};
    for (int i = 0; i < 10; ++i) {
        trace_k<<<2, 256, 0, stream>>>(cur, tr);
        float* dst = bufs[i & 1];
        wmma_gemm_f32<<<dim3(1024 / TILE_N, 1024 / TILE_M, 2), 256, 0, stream>>>(
            cur, cur, dst, 1024, 1024, 1024, MAT, MAT, MAT, tr, 2, nullptr);
        cur = dst;
    }

    // ---- [3] sigma via Rayleigh quotient; build epilogue scalars ----
    finalize_sigma<<<2, 1024, 0, stream>>>(cur, Gcat, log_gamma, scal);

    // ---- [4] u_scaled = (u @ W_in^T) * (exp(log_gamma)/max(sigma_in,1)) ----
    wmma_gemm_f32<<<dim3(Nn / TILE_N, BT / TILE_M, 1), 256, 0, stream>>>(
        u, W_in, us, BT, Nn, D, 0, 0, 0, scal + 0, 1, nullptr);

    // ---- [5] raw = u @ Wp^T + bp ----
    wmma_gemm_f32<<<dim3(3 * Nn / TILE_N, BT / TILE_M, 1), 256, 0, stream>>>(
        u, Wp, raw, BT, 3 * Nn, D, 0, 0, 0, nullptr, 0, bp);

    // ---- [6] elementwise a,b,c (+2x2 spectral clamp), in-place ----
    elemwise_abc<<<(long)BT * Nn / 256, 256, 0, stream>>>(raw, us, alpha_log, delta_bias);

    // ---- [7] chunked scan over T ----
    scan_pass1<<<dim3(Nn / 256, SCAN_NCHUNK, Bb), 256, 0, stream>>>(raw, Ac, Bc);
    scan_pass2<<<(Bb * Nn) / 256, 256, 0, stream>>>(Ac, Bc, Sst);
    scan_pass3<<<dim3(Nn / 256, SCAN_NCHUNK, Bb), 256, 0, stream>>>(raw, Sst, us);

    // ---- [8] out = (yhat @ W_out^T) * (1/max(sigma_out,1)) ----
    wmma_gemm_f32<<<dim3(Nn / TILE_N, BT / TILE_M, 1), 256, 0, stream>>>(
        us, W_out, out, BT, Nn, Nn, 0, 0, 0, scal + 1, 1, nullptr);
}